// TransactionClassifier_68839735820742
// MI455X (gfx1250) — compile-verified
//
#include <hip/hip_runtime.h>
#include <math.h>

#define N_NODES 50000
#define N_EDGES 600000
#define E_TOT   (N_EDGES + N_NODES)
#define F1      256      /* HEADS*HID */
#define HID     128
#define NEG_SLOPE 0.2f
#define K_MAX   256

typedef __attribute__((ext_vector_type(2))) float v2f;
typedef __attribute__((ext_vector_type(8))) float v8f;
typedef __attribute__((ext_vector_type(4))) int   v4i;

#if defined(__has_builtin)
#if __has_builtin(__builtin_amdgcn_global_load_async_to_lds_b128) && \
    __has_builtin(__builtin_amdgcn_s_wait_asynccnt)
#define HAVE_ASYNC_LDS 1
#endif
#endif

// ---------------------------------------------------------------------------
// float atomic max via int punning (nmax initialized to -inf; every node has a
// self-loop so the result is finite).
__device__ __forceinline__ void atomicMaxF32(float* addr, float v) {
  if (v >= 0.0f) atomicMax((int*)addr, __float_as_int(v));
  else           atomicMin((unsigned int*)addr, __float_as_uint(v));
}

// ---------------------------------------------------------------------------
// Y[m,n] = sum_k X[m,k]*W[k,n] + b[n]   (fp32, V_WMMA_F32_16X16X4_F32)
//
// Block = 4 waves (128 threads). Block tile = 64 rows x 64 cols.
// The K x 64 B-panel is staged in LDS once per block (async global->LDS when
// available). Each wave owns a 16 x 64 strip: 4 v8f accumulators; per K-step
// of 4 it does one b64 A load + 8 ds_load_b32 and issues 4 WMMAs.
//
// Fragment layouts (ISA 7.12.2, 32-bit):
//  A 16x4: lanes 0-15 -> M=lane, K={k,k+1}; lanes 16-31 -> M=lane-16, K={k+2,k+3}
//  B 4x16: VGPR r: lanes 0-15 K=k+r, lanes 16-31 K=k+r+2, N=lane%16
//  C/D 16x16: VGPR r: lanes 0-15 M=r, lanes 16-31 M=r+8, N=lane%16
__global__ __launch_bounds__(128)
void gemm_wmma_f32(const float* __restrict__ X, const float* __restrict__ W,
                   const float* __restrict__ bvec, float* __restrict__ Y,
                   int M, int K, int F) {
  __shared__ float Bs[K_MAX * 64];   // 64 KB max

  const int tid  = threadIdx.x;
  const int wave = tid >> 5;
  const int lane = tid & 31;
  const int half = lane >> 4;
  const int l16  = lane & 15;
  const int n0 = blockIdx.y * 64;
  const int m0 = blockIdx.x * 64 + wave * 16;

  // ---- stage B panel (K x 64) into LDS, cooperatively, 16B per transfer ----
#if HAVE_ASYNC_LDS
  for (int i = tid; i < K * 16; i += 128) {
    int k = i >> 4, n4 = (i & 15) << 2;
    __builtin_amdgcn_global_load_async_to_lds_b128(
        (__attribute__((address_space(1))) v4i*)(uintptr_t)&W[(size_t)k * F + n0 + n4],
        (__attribute__((address_space(3))) v4i*)&Bs[k * 64 + n4],
        0, 0);
  }
  __builtin_amdgcn_s_wait_asynccnt(0);
#else
  for (int i = tid; i < K * 16; i += 128) {
    int k = i >> 4, n4 = (i & 15) << 2;
    *(float4*)&Bs[k * 64 + n4] = *(const float4*)&W[(size_t)k * F + n0 + n4];
  }
#endif
  __syncthreads();

  if (m0 < M) {
    v8f acc[4];
#pragma unroll
    for (int t = 0; t < 4; ++t) acc[t] = (v8f){0.f,0.f,0.f,0.f,0.f,0.f,0.f,0.f};

    const float* xrow = X + (size_t)(m0 + l16) * K + 2 * half;
    for (int k = 0; k < K; k += 4) {
      v2f a = *(const v2f*)(xrow + k);               // one b64 load per lane
      const float* bs = &Bs[(k + 2 * half) * 64 + l16];
#pragma unroll
      for (int t = 0; t < 4; ++t) {
        v2f b;
        b.x = bs[t * 16];          // row k+2*half
        b.y = bs[64 + t * 16];     // row k+2*half+1
        acc[t] = __builtin_amdgcn_wmma_f32_16x16x4_f32(false, a, false, b,
                                                       (short)0, acc[t],
                                                       false, false);
      }
    }

#pragma unroll
    for (int t = 0; t < 4; ++t) {
      const float bv = bvec[n0 + t * 16 + l16];
#pragma unroll
      for (int r = 0; r < 8; ++r) {
        int row = m0 + r + half * 8;
        Y[(size_t)row * F + n0 + t * 16 + l16] = acc[t][r] + bv;
      }
    }
  }
}

// ---------------------------------------------------------------------------
__global__ void fill_f32(float* __restrict__ p, float v, int n) {
  int i = blockIdx.x * blockDim.x + threadIdx.x;
  if (i < n) p[i] = v;
}

// ---------------------------------------------------------------------------
// e[edge][h] = att[h] . leaky_relu(xs[src] + xd[dst]); segment-max into nmax.
// One wave per edge; lanes 0-15 -> head 0, lanes 16-31 -> head 1.
__global__ void edge_logits(const float* __restrict__ xs, const float* __restrict__ xd,
                            const long long* __restrict__ ei, const float* __restrict__ att,
                            float* __restrict__ elog, float* __restrict__ nmax) {
  int edge = blockIdx.x * (blockDim.x >> 5) + (threadIdx.x >> 5);
  if (edge >= E_TOT) return;
  int lane = threadIdx.x & 31;
  long long s, d;
  if (edge < N_EDGES) { s = ei[edge]; d = ei[N_EDGES + edge]; }
  else                { s = d = (long long)(edge - N_EDGES); }
  int head = lane >> 4;
  int l16  = lane & 15;
  const float* xsp = xs + (size_t)s * F1 + head * HID;
  const float* xdp = xd + (size_t)d * F1 + head * HID;
  const float* ap  = att + head * HID;
  float sum = 0.f;
#pragma unroll
  for (int i = 0; i < 8; ++i) {
    int c = l16 + 16 * i;
    float m = xsp[c] + xdp[c];
    m = (m > 0.f) ? m : NEG_SLOPE * m;
    sum += ap[c] * m;
  }
  sum += __shfl_xor(sum, 8, 32);
  sum += __shfl_xor(sum, 4, 32);
  sum += __shfl_xor(sum, 2, 32);
  sum += __shfl_xor(sum, 1, 32);
  if (l16 == 0) {
    elog[edge * 2 + head] = sum;
    atomicMaxF32(&nmax[d * 2 + head], sum);
  }
}

// ---------------------------------------------------------------------------
// elog <- exp(elog - nmax[dst]); segment-sum into nsum. One thread per (edge,head).
__global__ void edge_expsum(const long long* __restrict__ ei, const float* __restrict__ nmax,
                            float* __restrict__ elog, float* __restrict__ nsum) {
  int idx = blockIdx.x * blockDim.x + threadIdx.x;
  if (idx >= E_TOT * 2) return;
  int e = idx >> 1, h = idx & 1;
  long long d = (e < N_EDGES) ? ei[N_EDGES + e] : (long long)(e - N_EDGES);
  float v = __expf(elog[idx] - nmax[d * 2 + h]);
  elog[idx] = v;
  atomicAdd(&nsum[d * 2 + h], v);
}

// ---------------------------------------------------------------------------
// out[dst] += alpha * xs[src], per head. One wave per edge, 8 channels/lane.
__global__ void edge_aggregate(const float* __restrict__ xs, const long long* __restrict__ ei,
                               const float* __restrict__ elog, const float* __restrict__ nsum,
                               float* __restrict__ outb) {
  int edge = blockIdx.x * (blockDim.x >> 5) + (threadIdx.x >> 5);
  if (edge >= E_TOT) return;
  int lane = threadIdx.x & 31;
  long long s, d;
  if (edge < N_EDGES) { s = ei[edge]; d = ei[N_EDGES + edge]; }
  else                { s = d = (long long)(edge - N_EDGES); }
  float a0 = elog[edge * 2 + 0] / (nsum[d * 2 + 0] + 1e-16f);
  float a1 = elog[edge * 2 + 1] / (nsum[d * 2 + 1] + 1e-16f);
  const float* xrow = xs + (size_t)s * F1;
  float* orow = outb + (size_t)d * F1;
#pragma unroll
  for (int i = 0; i < 8; ++i) {
    int c = lane + 32 * i;
    float alpha = (c < HID) ? a0 : a1;
    atomicAdd(&orow[c], alpha * xrow[c]);
  }
}

// ---------------------------------------------------------------------------
__global__ void bias_relu(float* __restrict__ h, const float* __restrict__ bias, int n) {
  int i = blockIdx.x * blockDim.x + threadIdx.x;
  if (i >= n) return;
  float v = h[i] + bias[i & (F1 - 1)];
  h[i] = v > 0.f ? v : 0.f;
}

// ---------------------------------------------------------------------------
// y[node] = sigmoid(h3[node] . W4 + b4). One wave per node.
__global__ void post_head(const float* __restrict__ h3, const float* __restrict__ W4,
                          const float* __restrict__ b4, float* __restrict__ y) {
  int node = blockIdx.x * (blockDim.x >> 5) + (threadIdx.x >> 5);
  if (node >= N_NODES) return;
  int lane = threadIdx.x & 31;
  const float* row = h3 + (size_t)node * HID;
  float s = 0.f;
#pragma unroll
  for (int i = 0; i < 4; ++i) { int c = lane + 32 * i; s += row[c] * W4[c]; }
  s += __shfl_xor(s, 16, 32);
  s += __shfl_xor(s,  8, 32);
  s += __shfl_xor(s,  4, 32);
  s += __shfl_xor(s,  2, 32);
  s += __shfl_xor(s,  1, 32);
  if (lane == 0) y[node] = 1.0f / (1.0f + __expf(-(s + b4[0])));
}

// ---------------------------------------------------------------------------
static void run_layer(const float* h_in, int Kin,
                      const float* Wl, const float* bl, const float* Wr, const float* br,
                      const float* att, const float* bias, const long long* ei,
                      float* xs, float* xd, float* outb, float* elog,
                      float* nmax, float* nsum, hipStream_t stream) {
  dim3 gg((N_NODES + 63) / 64, F1 / 64);
  gemm_wmma_f32<<<gg, 128, 0, stream>>>(h_in, Wl, bl, xs, N_NODES, Kin, F1);
  gemm_wmma_f32<<<gg, 128, 0, stream>>>(h_in, Wr, br, xd, N_NODES, Kin, F1);
  fill_f32<<<(N_NODES * 2 + 255) / 256, 256, 0, stream>>>(nmax, -INFINITY, N_NODES * 2);
  fill_f32<<<(N_NODES * 2 + 255) / 256, 256, 0, stream>>>(nsum, 0.f, N_NODES * 2);
  fill_f32<<<(N_NODES * F1 + 255) / 256, 256, 0, stream>>>(outb, 0.f, N_NODES * F1);
  edge_logits<<<(E_TOT + 7) / 8, 256, 0, stream>>>(xs, xd, ei, att, elog, nmax);
  edge_expsum<<<(E_TOT * 2 + 255) / 256, 256, 0, stream>>>(ei, nmax, elog, nsum);
  edge_aggregate<<<(E_TOT + 7) / 8, 256, 0, stream>>>(xs, ei, elog, nsum, outb);
  bias_relu<<<(N_NODES * F1 + 255) / 256, 256, 0, stream>>>(outb, bias, N_NODES * F1);
}

extern "C" void kernel_launch(void* const* d_in, const int* in_sizes, int n_in,
                              void* d_out, int out_size, void* d_ws, size_t ws_size,
                              hipStream_t stream) {
  const float*     x    = (const float*)d_in[0];
  const long long* ei   = (const long long*)d_in[1];   // int64 edge_index [2,E]
  const float* Wl1  = (const float*)d_in[2],  *bl1   = (const float*)d_in[3];
  const float* Wr1  = (const float*)d_in[4],  *br1   = (const float*)d_in[5];
  const float* att1 = (const float*)d_in[6],  *bias1 = (const float*)d_in[7];
  const float* Wl2  = (const float*)d_in[8],  *bl2   = (const float*)d_in[9];
  const float* Wr2  = (const float*)d_in[10], *br2   = (const float*)d_in[11];
  const float* att2 = (const float*)d_in[12], *bias2 = (const float*)d_in[13];
  const float* W3   = (const float*)d_in[14], *b3    = (const float*)d_in[15];
  const float* W4   = (const float*)d_in[16], *b4    = (const float*)d_in[17];

  float* ws   = (float*)d_ws;
  float* xs   = ws;                               // N*256
  float* xd   = xs   + (size_t)N_NODES * F1;      // N*256
  float* outb = xd   + (size_t)N_NODES * F1;      // N*256 (layer output / next input)
  float* elog = outb + (size_t)N_NODES * F1;      // E_TOT*2
  float* nmax = elog + (size_t)E_TOT * 2;         // N*2
  float* nsum = nmax + (size_t)N_NODES * 2;       // N*2

  // Layer 1: x (N x 128) -> outb = relu(gatv2 + bias1)   (N x 256)
  run_layer(x, 128, Wl1, bl1, Wr1, br1, att1, bias1, ei,
            xs, xd, outb, elog, nmax, nsum, stream);
  // Layer 2: outb (N x 256) -> outb = relu(gatv2 + bias2) (N x 256)
  run_layer(outb, 256, Wl2, bl2, Wr2, br2, att2, bias2, ei,
            xs, xd, outb, elog, nmax, nsum, stream);

  // post_mp: h3 = outb @ W3 + b3  (N x 128), stored in xs (reused)
  dim3 g3((N_NODES + 63) / 64, HID / 64);
  gemm_wmma_f32<<<g3, 128, 0, stream>>>(outb, W3, b3, xs, N_NODES, F1, HID);
  // y = sigmoid(h3 @ W4 + b4)
  post_head<<<(N_NODES + 7) / 8, 256, 0, stream>>>(xs, W4, b4, (float*)d_out);
}